// LengthAdaptor_20993800142993
// MI455X (gfx1250) — compile-verified
//
#include <hip/hip_runtime.h>

// ---------------- problem constants (match reference) ----------------
#define B_    32
#define S_    1024
#define D_    256
#define F2_   256
#define T_    8192
#define KT_   192          // 3 taps * (256/4) k-chunks of 4
#define LN_EPS 1e-5f
// LDS row stride in floats: 272*4 bytes = 1088 (16B aligned for async b128),
// and (1088/4) % 64 == 16  ->  bank = (16*row + col) % 64: rows 4 apart share banks in the
// 16-row ds_load_b64 A-fragment pattern.
#define XS_STRIDE 272

typedef __attribute__((ext_vector_type(2))) float v2f;
typedef __attribute__((ext_vector_type(4))) float v4f;
typedef __attribute__((ext_vector_type(8))) float v8f;

// ---------------------------------------------------------------------
// Reorder conv weight w[F][C][3] into WMMA B-fragment order:
//   wr[((nt*KT + kt)*32 + lane)*2 + q] = w[nt*16 + lane%16][ (kt%64)*4 + (lane/16)*2 + q ][ kt/64 ]
// so each WMMA step loads one coalesced float2 per lane.
// ---------------------------------------------------------------------
__global__ __launch_bounds__(256)
void reorder_w_kernel(const float* __restrict__ w, float* __restrict__ wr) {
    int idx = blockIdx.x * 256 + threadIdx.x;
    if (idx >= F2_ * D_ * 3) return;
    int q    = idx & 1;
    int lane = (idx >> 1) & 31;
    int kt   = (idx >> 6) % KT_;
    int nt   = idx / (KT_ * 64);
    int n    = nt * 16 + (lane & 15);
    int tap  = kt >> 6;
    int c    = ((kt & 63) << 2) + ((lane >> 4) << 1) + q;
    wr[idx] = w[(n * D_ + c) * 3 + tap];
}

// ---------------------------------------------------------------------
// Fused conv1d(K=3, pad=1) + bias + ReLU + LayerNorm.
// mode 0: write normalized tile to h_out [B,S,256]
// mode 1: additionally fold in the final linear (h @ lin_w + lin_b) and the
//         mask, writing log_dur [B,S]; h itself is not materialized.
// Block = 256 threads (8 waves); block computes a [16 x 256] output tile.
// x tile staged with GLOBAL_LOAD_ASYNC_TO_LDS_B128 (ASYNCcnt path),
// K=768 accumulated via 192 x v_wmma_f32_16x16x4_f32 per accumulator.
// ---------------------------------------------------------------------
__global__ __launch_bounds__(256)
void conv_ln_kernel(const float* __restrict__ in,      // [B,S,256]
                    const float* __restrict__ wr,      // reordered weights
                    const float* __restrict__ bias,    // [256]
                    const float* __restrict__ gamma,   // [256]
                    const float* __restrict__ beta,    // [256]
                    float* __restrict__ h_out,         // mode 0
                    const float* __restrict__ lin_w,   // mode 1
                    const float* __restrict__ lin_b,   // mode 1
                    const unsigned char* __restrict__ mask, // mode 1, [B,S] bool
                    float* __restrict__ logd,          // mode 1, [B,S]
                    int mode) {
    __shared__ float smem[18 * XS_STRIDE];             // 18 halo'd x rows, reused for h tile

    const int b   = blockIdx.x >> 6;
    const int s0  = (blockIdx.x & 63) << 4;
    const int tid = threadIdx.x;

    // ---- zero out-of-range halo rows (only first/last tile of each batch) ----
    if (s0 == 0 && tid < 256) {
        smem[0 * XS_STRIDE + tid] = 0.f;               // row 0 <- s = -1
    }
    if (s0 == S_ - 16 && tid < 256) {
        smem[17 * XS_STRIDE + tid] = 0.f;              // row 17 <- s = S
    }

    // ---- async-stage rows s0-1 .. s0+16 of input directly into LDS ----
    for (int i = tid; i < 18 * 64; i += 256) {
        int row = i >> 6;
        int c4  = (i & 63) << 2;
        int s   = s0 - 1 + row;
        if (s >= 0 && s < S_) {
            unsigned lds = (unsigned)(unsigned long long)&smem[row * XS_STRIDE + c4];
            unsigned long long gaddr =
                (unsigned long long)(in + ((size_t)(b * S_ + s)) * D_ + c4);
            asm volatile("global_load_async_to_lds_b128 %0, %1, off"
                         :: "v"(lds), "v"(gaddr) : "memory");
        }
    }
    asm volatile("s_wait_asynccnt 0x0" ::: "memory");
    __syncthreads();

    const int wave = tid >> 5;
    const int lane = tid & 31;
    const int half = lane >> 4;
    const int l16  = lane & 15;
    const int nt0  = wave * 2;
    const int nt1  = wave * 2 + 1;

    v8f acc0 = {0.f,0.f,0.f,0.f,0.f,0.f,0.f,0.f};
    v8f acc1 = acc0;

    const float* wr0 = wr + (size_t)nt0 * KT_ * 64;
    const float* wr1 = wr + (size_t)nt1 * KT_ * 64;

    for (int kt = 0; kt < KT_; ++kt) {
        const int tap = kt >> 6;
        const int c   = ((kt & 63) << 2) + (half << 1);
        // A fragment: rows striped over lanes, K pair per lane -> ds_load_b64
        v2f a = *(const v2f*)&smem[(l16 + tap) * XS_STRIDE + c];
        // B fragments: reordered, lane-contiguous -> coalesced global b64 (L2-hot)
        v2f b0 = *(const v2f*)(wr0 + ((kt << 5) + lane) * 2);
        v2f b1 = *(const v2f*)(wr1 + ((kt << 5) + lane) * 2);
        acc0 = __builtin_amdgcn_wmma_f32_16x16x4_f32(false, a, false, b0,
                                                     (short)0, acc0, false, false);
        acc1 = __builtin_amdgcn_wmma_f32_16x16x4_f32(false, a, false, b1,
                                                     (short)0, acc1, false, false);
    }
    __syncthreads();   // everyone done reading x tile; reuse smem for h tile

    // ---- bias + ReLU, scatter D fragments to LDS h tile [16][XS_STRIDE] ----
    {
        const float bv0 = bias[nt0 * 16 + l16];
        const float bv1 = bias[nt1 * 16 + l16];
        #pragma unroll
        for (int r = 0; r < 8; ++r) {
            const int row = r + half * 8;
            float y0 = acc0[r] + bv0; y0 = y0 > 0.f ? y0 : 0.f;
            float y1 = acc1[r] + bv1; y1 = y1 > 0.f ? y1 : 0.f;
            smem[row * XS_STRIDE + nt0 * 16 + l16] = y0;
            smem[row * XS_STRIDE + nt1 * 16 + l16] = y1;
        }
    }
    __syncthreads();

    // ---- LayerNorm over 256 channels; each wave handles 2 rows ----
    for (int rr = wave * 2; rr < wave * 2 + 2; ++rr) {
        const int s = s0 + rr;
        float v[8];
        float sum = 0.f, sq = 0.f;
        #pragma unroll
        for (int i = 0; i < 8; ++i) {
            v[i] = smem[rr * XS_STRIDE + lane + i * 32];
            sum += v[i];
            sq  += v[i] * v[i];
        }
        #pragma unroll
        for (int off = 16; off > 0; off >>= 1) {
            sum += __shfl_xor(sum, off, 32);
            sq  += __shfl_xor(sq,  off, 32);
        }
        const float mu   = sum * (1.0f / 256.0f);
        const float var  = sq * (1.0f / 256.0f) - mu * mu;
        const float rstd = rsqrtf(var + LN_EPS);

        if (mode == 0) {
            #pragma unroll
            for (int i = 0; i < 8; ++i) {
                const int ch = lane + i * 32;                // coalesced stores
                h_out[((size_t)(b * S_ + s)) * D_ + ch] =
                    (v[i] - mu) * rstd * gamma[ch] + beta[ch];
            }
        } else {
            float part = 0.f;
            #pragma unroll
            for (int i = 0; i < 8; ++i) {
                const int ch = lane + i * 32;
                const float y = (v[i] - mu) * rstd * gamma[ch] + beta[ch];
                part += y * lin_w[ch];
            }
            #pragma unroll
            for (int off = 16; off > 0; off >>= 1)
                part += __shfl_xor(part, off, 32);
            if (lane == 0) {
                float ld = part + lin_b[0];
                if (mask[b * S_ + s]) ld = 0.0f;
                logd[b * S_ + s] = ld;
            }
        }
    }
}

// ---------------------------------------------------------------------
// Per-batch inclusive scan of durations (one block per batch row).
// Also emits dur (as float) and mel_len = min(total, T).
// ---------------------------------------------------------------------
__global__ __launch_bounds__(256)
void scan_kernel(const int* __restrict__ dur, int* __restrict__ cum,
                 float* __restrict__ dur_out, float* __restrict__ mel_out) {
    __shared__ int part[256];
    const int b   = blockIdx.x;
    const int tid = threadIdx.x;

    int4 d = *(const int4*)(dur + b * S_ + tid * 4);
    const int l0 = d.x;
    const int l1 = l0 + d.y;
    const int l2 = l1 + d.z;
    const int l3 = l2 + d.w;
    part[tid] = l3;
    __syncthreads();

    for (int off = 1; off < 256; off <<= 1) {
        int t = (tid >= off) ? part[tid - off] : 0;
        __syncthreads();
        part[tid] += t;
        __syncthreads();
    }
    const int excl = part[tid] - l3;

    int4 c; c.x = excl + l0; c.y = excl + l1; c.z = excl + l2; c.w = excl + l3;
    *(int4*)(cum + b * S_ + tid * 4) = c;

    float4 df = make_float4((float)d.x, (float)d.y, (float)d.z, (float)d.w);
    *(float4*)(dur_out + b * S_ + tid * 4) = df;

    if (tid == 255) {
        const int tot = part[255];
        mel_out[b] = (float)(tot < T_ ? tot : T_);
    }
}

// ---------------------------------------------------------------------
// Length regulator: one wave per output frame t. Wave-uniform binary
// search (searchsorted right) over the L2-resident cum row, then stream
// one 1 KB source row with 2x b128 loads + 2x NON-TEMPORAL b128 stores
// per lane. NT stores keep x (32 MB) and cum hot in the 192 MB L2 while
// the 268 MB write-once stream goes straight to HBM (~11.5 us floor).
// ---------------------------------------------------------------------
__global__ __launch_bounds__(256)
void gather_kernel(const float* __restrict__ x, const int* __restrict__ cum,
                   float* __restrict__ out) {
    const int gw   = blockIdx.x * 8 + (threadIdx.x >> 5);   // global wave = b*T + t
    const int lane = threadIdx.x & 31;
    const int b = gw >> 13;                                  // T = 8192
    const int t = gw & (T_ - 1);

    const int* crow = cum + b * S_;
    int mel = crow[S_ - 1];
    if (mel > T_) mel = T_;

    const size_t obase = ((size_t)b * T_ + t) * (size_t)D_;

    if (t >= mel) {
        const v4f z = {0.f, 0.f, 0.f, 0.f};
        __builtin_nontemporal_store(z, (v4f*)(out + obase + lane * 4));
        __builtin_nontemporal_store(z, (v4f*)(out + obase + 128 + lane * 4));
        return;
    }

    int lo = 0, hi = S_;
    while (lo < hi) {
        const int mid = (lo + hi) >> 1;
        if (crow[mid] <= t) lo = mid + 1; else hi = mid;
    }
    if (lo > S_ - 1) lo = S_ - 1;

    const float* src = x + ((size_t)b * S_ + lo) * (size_t)D_;
    v4f a0 = *(const v4f*)(src + lane * 4);
    v4f a1 = *(const v4f*)(src + 128 + lane * 4);
    __builtin_nontemporal_store(a0, (v4f*)(out + obase + lane * 4));
    __builtin_nontemporal_store(a1, (v4f*)(out + obase + 128 + lane * 4));
}

// ---------------------------------------------------------------------
extern "C" void kernel_launch(void* const* d_in, const int* in_sizes, int n_in,
                              void* d_out, int out_size, void* d_ws, size_t ws_size,
                              hipStream_t stream) {
    (void)in_sizes; (void)n_in; (void)out_size; (void)ws_size;

    const float*         x        = (const float*)d_in[0];
    const unsigned char* mask     = (const unsigned char*)d_in[1];
    const int*           duration = (const int*)d_in[2];
    /* d_in[3] = max_len scalar (compile-time T_) */
    const float* conv1_w = (const float*)d_in[4];
    const float* conv1_b = (const float*)d_in[5];
    const float* ln1_g   = (const float*)d_in[6];
    const float* ln1_b   = (const float*)d_in[7];
    const float* conv2_w = (const float*)d_in[8];
    const float* conv2_b = (const float*)d_in[9];
    const float* ln2_g   = (const float*)d_in[10];
    const float* ln2_b   = (const float*)d_in[11];
    const float* lin_w   = (const float*)d_in[12];
    const float* lin_b   = (const float*)d_in[13];

    // workspace layout (floats)
    const size_t WSZ = (size_t)F2_ * D_ * 3;          // 196608 per conv weight
    float* ws  = (float*)d_ws;
    float* W1R = ws;
    float* W2R = ws + WSZ;
    float* H1  = ws + 2 * WSZ;                        // [B,S,256]
    int*   CUM = (int*)(ws + 2 * WSZ + (size_t)B_ * S_ * D_);

    // output layout (floats): out | log_dur | dur | mel_len
    float* out   = (float*)d_out;
    float* logd  = out + (size_t)B_ * T_ * D_;
    float* dur_o = logd + (size_t)B_ * S_;
    float* mel_o = dur_o + (size_t)B_ * S_;

    const int RB = (int)((WSZ + 255) / 256);
    reorder_w_kernel<<<RB, 256, 0, stream>>>(conv1_w, W1R);
    reorder_w_kernel<<<RB, 256, 0, stream>>>(conv2_w, W2R);

    conv_ln_kernel<<<B_ * (S_ / 16), 256, 0, stream>>>(
        x, W1R, conv1_b, ln1_g, ln1_b, H1,
        nullptr, nullptr, nullptr, nullptr, 0);

    conv_ln_kernel<<<B_ * (S_ / 16), 256, 0, stream>>>(
        H1, W2R, conv2_b, ln2_g, ln2_b, nullptr,
        lin_w, lin_b, mask, logd, 1);

    scan_kernel<<<B_, 256, 0, stream>>>(duration, CUM, dur_o, mel_o);

    gather_kernel<<<(B_ * T_) / 8, 256, 0, stream>>>(x, CUM, out);
}